// ParametricKernelLayer_81406810128832
// MI455X (gfx1250) — compile-verified
//
#include <hip/hip_runtime.h>
#include <hip/hip_bf16.h>

// Problem constants (from reference)
#define NF 8
#define NK 16
#define NY 16
#define NX 16
#define NP 32
#define NT 8
#define NB 64
#define SDIM (NY*NX*NP*NT)   // 65536
#define CHUNK 128            // s-values staged per iteration (shares fixed y,x)
#define PITCH 132            // LDS row pitch (multiple of 4 for b128 async stores)
#define SPLIT 32             // s-dim blocks per field f

typedef float v2f __attribute__((ext_vector_type(2)));
typedef float v8f __attribute__((ext_vector_type(8)));

// ---------------------------------------------------------------------------
// Zero the output (we accumulate with atomics; harness poisons d_out).
// ---------------------------------------------------------------------------
__global__ void pk_zero(float* __restrict__ out, int n) {
    int i = blockIdx.x * 256 + threadIdx.x;
    if (i < n) out[i] = 0.0f;
}

// ---------------------------------------------------------------------------
// Precompute separable 1-D kernel factors and fold in 1/integrated.
// One thread per (f,k) pair; 128 threads total.
// ws layout (floats): [0,2048)  wlat[f*16+k][16]  (scaled by 1/integrated)
//                     [2048,..) wlon[f*16+k][16]
//                     [4096,..) wlev[f*16+k][32]
//                     [8192,..) wtime[f*16+k][8]
// ---------------------------------------------------------------------------
__global__ __launch_bounds__(128) void pk_precompute(
    const float* __restrict__ qw,
    const float* __restrict__ mu_lat, const float* __restrict__ ls_lat,
    const float* __restrict__ mu_lon, const float* __restrict__ ls_lon,
    const float* __restrict__ mu_lev, const float* __restrict__ ls_lev,
    const float* __restrict__ logtau,
    float* __restrict__ w)
{
    __shared__ float lat[128][NY];
    __shared__ float lon[128][NX];
    __shared__ float lev[128][NP];
    __shared__ float tim[128][NT];
    const int t = threadIdx.x;   // t = f*16 + k

    {
        float mu = mu_lat[t], sg = __expf(ls_lat[t]);
        #pragma unroll
        for (int y = 0; y < NY; ++y) {
            float c = -1.0f + (2.0f / 15.0f) * (float)y;
            float z = (c - mu) / sg;
            lat[t][y] = __expf(-0.5f * z * z);
        }
    }
    {
        float mu = mu_lon[t], sg = __expf(ls_lon[t]);
        #pragma unroll
        for (int x = 0; x < NX; ++x) {
            float c = -1.0f + (2.0f / 15.0f) * (float)x;
            float z = (c - mu) / sg;
            lon[t][x] = __expf(-0.5f * z * z);
        }
    }
    {
        float mu = mu_lev[t], sg = __expf(ls_lev[t]);
        #pragma unroll
        for (int p = 0; p < NP; ++p) {
            float c = -1.0f + (2.0f / 31.0f) * (float)p;
            float z = (c - mu) / sg;
            lev[t][p] = __expf(-0.5f * z * z);
        }
    }
    {
        float tau = __expf(logtau[t]) + 1e-6f;
        #pragma unroll
        for (int tt = 0; tt < NT; ++tt)
            tim[t][tt] = __expf(-(float)tt / tau);
    }

    // integrated[f,k] = sum_{y,x,p,t} klat*klon*klev*ktime*qw
    float acc = 0.0f;
    const float* q = qw;
    for (int y = 0; y < NY; ++y) {
        float a = lat[t][y];
        for (int x = 0; x < NX; ++x) {
            float axy = a * lon[t][x];
            for (int p = 0; p < NP; ++p) {
                float axyp = axy * lev[t][p];
                #pragma unroll
                for (int tt = 0; tt < NT; ++tt)
                    acc += axyp * tim[t][tt] * q[tt];
                q += NT;
            }
        }
    }
    float inv = 1.0f / (acc + 1e-4f);

    #pragma unroll
    for (int y = 0; y < NY; ++y) w[t * 16 + y] = lat[t][y] * inv;
    #pragma unroll
    for (int x = 0; x < NX; ++x) w[2048 + t * 16 + x] = lon[t][x];
    #pragma unroll
    for (int p = 0; p < NP; ++p) w[4096 + t * 32 + p] = lev[t][p];
    #pragma unroll
    for (int tt = 0; tt < NT; ++tt) w[8192 + t * 8 + tt] = tim[t][tt];
}

// ---------------------------------------------------------------------------
// Main GEMM: features[b, f*16+k] += sum_s patch[b,f,s] * W[f,k,s]
// grid = (NF, SPLIT), block = 256 (8 waves).
// Double-buffered pipeline: next chunk's patch tile is DMA'd into LDS with
// GLOBAL_LOAD_ASYNC_TO_LDS_B128 (ASYNCcnt) while the current chunk runs
// 16 x V_WMMA_F32_16X16X4_F32 per wave.
// ---------------------------------------------------------------------------
__global__ __launch_bounds__(256) void pk_main(
    const float* __restrict__ patch,
    const float* __restrict__ qw,
    const float* __restrict__ w,
    float* __restrict__ out)
{
    __shared__ float pt[2][NB * PITCH];   // 2 x (64 x 128) patch tiles
    __shared__ float wt[2][NK * PITCH];   // 2 x (16 x 128) weight tiles

    const int f    = blockIdx.x;
    const int sblk = blockIdx.y;
    const int tid  = threadIdx.x;
    const int lane = tid & 31;
    const int wv   = tid >> 5;
    const int mtile = wv >> 1;        // 0..3: batch rows [mtile*16, +16)
    const int shalf = wv & 1;         // 0..1: s-half of the chunk
    const int h    = lane >> 4;       // lane half (selects K pair / M+8)
    const int n    = lane & 15;       // row (A) / column (B,C)

    v8f c = {0, 0, 0, 0, 0, 0, 0, 0};

    const int chunks = SDIM / SPLIT / CHUNK;   // 16

    // Stage one chunk: async-DMA the patch tile into LDS, compute the weight
    // tile from the 1-D separable factors with normal LDS stores.
    auto stage = [&](int ci, int buf) {
        const int s_base = (sblk * chunks + ci) * CHUNK;

        // patch tile: 64 rows x 128 s-values; 8 async b128 per wave,
        // each lane moves 16B global -> LDS with no VGPR round trip.
        #pragma unroll
        for (int i = 0; i < 8; ++i) {
            int idx = tid + i * 256;          // 0..2047
            int row = idx >> 5;               // 0..63 (batch b)
            int c4  = idx & 31;               // float4 column
            const float* g = patch + ((size_t)row * NF + f) * SDIM + s_base + c4 * 4;
            // Low 32 bits of a generic __shared__ pointer are the LDS byte
            // address (ISA 10.2).
            unsigned ldsa = (unsigned)(uintptr_t)&pt[buf][row * PITCH + c4 * 4];
            asm volatile("global_load_async_to_lds_b128 %0, %1, off"
                         :: "v"(ldsa), "v"(g) : "memory");
        }

        // weight tile: W[f,k,s]*qw[s] from the 1-D factors (y,x fixed per chunk)
        int yx   = s_base >> 8;
        int y    = yx >> 4;
        int x    = yx & 15;
        int half = (s_base >> 7) & 1;
        #pragma unroll
        for (int i = 0; i < 8; ++i) {
            int idx = tid + i * 256;      // 0..2047
            int k   = idx >> 7;           // 0..15
            int j   = idx & 127;          // s within chunk
            int p   = half * 16 + (j >> 3);
            int tt  = j & 7;
            int fk  = f * 16 + k;
            float wv_ = w[fk * 16 + y] * w[2048 + fk * 16 + x] *
                        w[4096 + fk * 32 + p] * w[8192 + fk * 8 + tt] *
                        qw[s_base + j];
            wt[buf][k * PITCH + j] = wv_;
        }
    };

    stage(0, 0);   // prologue: fill buffer 0

    for (int ci = 0; ci < chunks; ++ci) {
        const int cur = ci & 1;

        if (ci)
            __syncthreads();              // barrier A: compute(ci-1) done ->
                                          // safe to overwrite buf[1-cur]
        if (ci + 1 < chunks) {
            stage(ci + 1, cur ^ 1);       // prefetch next chunk (async)
            // retire previous chunk's 8 async loads (in-order), leave the 8
            // just-issued ones in flight.
            asm volatile("s_wait_asynccnt 0x8" ::: "memory");
        } else {
            asm volatile("s_wait_asynccnt 0x0" ::: "memory");
        }
        __syncthreads();                  // barrier B: buf[cur] complete

        // --- 16 x WMMA f32 16x16x4 over this wave's 64-wide s-half ---
        // A layout: lane n holds row M=n; VGPR v, half h -> K = v + 2h.
        // B layout: lane n holds col N=n; VGPR v, half h -> K = v + 2h.
        const float* pA = &pt[cur][(mtile * 16 + n) * PITCH + shalf * 64 + 2 * h];
        const float* pB = &wt[cur][n * PITCH + shalf * 64 + 2 * h];
        #pragma unroll
        for (int j = 0; j < 16; ++j) {
            v2f a = *(const v2f*)(pA + j * 4);
            v2f b = *(const v2f*)(pB + j * 4);
            c = __builtin_amdgcn_wmma_f32_16x16x4_f32(
                    false, a, false, b, (short)0, c, false, false);
        }
    }

    // --- accumulate partial C into out[b, f*16+k] ---
    // C layout: VGPR r, lane half h -> M = r + 8h; N = lane & 15.
    #pragma unroll
    for (int r = 0; r < 8; ++r) {
        int b = mtile * 16 + r + 8 * h;
        atomicAdd(&out[b * (NF * NK) + f * NK + n], c[r]);
    }
}

// ---------------------------------------------------------------------------
extern "C" void kernel_launch(void* const* d_in, const int* in_sizes, int n_in,
                              void* d_out, int out_size, void* d_ws, size_t ws_size,
                              hipStream_t stream) {
    const float* patch  = (const float*)d_in[0];
    const float* qw     = (const float*)d_in[1];
    const float* mu_lat = (const float*)d_in[2];
    const float* ls_lat = (const float*)d_in[3];
    const float* mu_lon = (const float*)d_in[4];
    const float* ls_lon = (const float*)d_in[5];
    const float* mu_lev = (const float*)d_in[6];
    const float* ls_lev = (const float*)d_in[7];
    const float* logtau = (const float*)d_in[8];
    float* out = (float*)d_out;          // 64 x 128 = 8192 floats
    float* w   = (float*)d_ws;           // 9216 floats (36 KB) of 1-D factors

    pk_zero<<<(NB * NF * NK + 255) / 256, 256, 0, stream>>>(out, NB * NF * NK);
    pk_precompute<<<1, 128, 0, stream>>>(qw, mu_lat, ls_lat, mu_lon, ls_lon,
                                         mu_lev, ls_lev, logtau, w);
    pk_main<<<dim3(NF, SPLIT), 256, 0, stream>>>(patch, qw, w, out);
}